// DMFNet_80307298500833
// MI455X (gfx1250) — compile-verified
//
#include <hip/hip_runtime.h>

// ---------------------------------------------------------------------------
// CDNA5 fp32 WMMA helpers (V_WMMA_F32_16X16X4_F32, wave32)
// A: 16x4 f32 (2 VGPR)  lanes0-15: M=lane,K=0/1 ; lanes16-31: M=lane-16,K=2/3
// B: 4x16 f32 (2 VGPR)  lanes0-15: N=lane,K=0/1 ; lanes16-31: N=lane-16,K=2/3
// C/D: 16x16 f32 (8 VGPR) vgpr r: M=r (lanes0-15) / M=r+8 (lanes16-31), N=lane%16
// ---------------------------------------------------------------------------
typedef float v2f __attribute__((ext_vector_type(2)));
typedef float v8f __attribute__((ext_vector_type(8)));

__device__ __forceinline__ v8f wmma_f32_k4(v2f a, v2f b, v8f c) {
  return __builtin_amdgcn_wmma_f32_16x16x4_f32(false, a, false, b, (short)0, c,
                                               false, false);
}

#define HWC 16384
#define WDIM 128

// CDNA5 async global->LDS copy (GLOBAL_LOAD_ASYNC_TO_LDS_B128, ASYNCcnt).
// Set to 0 to fall back to plain vectorized LDS stores.
#ifndef USE_ASYNC_LDS
#define USE_ASYNC_LDS 1
#endif

// ---------------------------------------------------------------------------
// Weight prep: We[o,c] = W[o,c] * si[c];  be[o] = bias[o] + sum_c W[o,c]*bi[c]
// Folds an *input-side* affine (aff -> conv) into the GEMM.
// ---------------------------------------------------------------------------
__global__ __launch_bounds__(256) void prep_weight(
    const float* __restrict__ W, const float* __restrict__ bias,
    const float* __restrict__ si, const float* __restrict__ bi,
    float* __restrict__ We, float* __restrict__ be, int Cin) {
  const int o = blockIdx.x;
  const int t = threadIdx.x;
  __shared__ float red[256];
  float acc = 0.f;
  for (int c = t; c < Cin; c += 256) {
    float w = W[o * Cin + c];
    We[o * Cin + c] = w * (si ? si[c] : 1.0f);
    if (bi) acc += w * bi[c];
  }
  red[t] = acc;
  __syncthreads();
  for (int s = 128; s > 0; s >>= 1) {
    if (t < s) red[t] += red[t + s];
    __syncthreads();
  }
  if (t == 0) be[o] = red[0] + (bias ? bias[o] : 0.0f);
}

// ---------------------------------------------------------------------------
// Fused 1x1-conv GEMM:  Y(MxN) = epilogue( We(MxCin) * X(CinxN) + be )
// X given as up to 4 segments of 64 channels each (for the concat GEMM).
// The Cin x NWG activation tile is staged in LDS via async b128 copies;
// WMMA B-fragments are then read from LDS, A-fragments (weights) from L1.
// Epilogue: +be ; [y=s1*y+b1] ; [relu] ; [y=s2*y+b2 ; relu] ; [y=relu(y+res)]
// 256 thr = 8 waves; M=64 -> 4x2 tiles (NWG=32), M=16 -> 1x8 tiles (NWG=128)
// ---------------------------------------------------------------------------
__global__ __launch_bounds__(256) void conv1x1_gemm(
    const float* __restrict__ X0, const float* __restrict__ X1,
    const float* __restrict__ X2, const float* __restrict__ X3, int Cin, int M,
    const float* __restrict__ We, const float* __restrict__ be,
    const float* __restrict__ s1, const float* __restrict__ b1, int relu1,
    const float* __restrict__ s2, const float* __restrict__ b2,
    const float* __restrict__ resadd, float* __restrict__ out) {
  __shared__ __align__(16) float Xs[9216];  // max(256*36, 64*132) floats
  const int b = blockIdx.y;
  const int tid = threadIdx.x;
  const int wave = tid >> 5, lane = tid & 31, half = lane >> 4, ln = lane & 15;
  const int tilesM = M >> 4;
  const int tilesNwg = 8 / tilesM;
  const int NWG = tilesNwg * 16;   // columns per workgroup
  const int NWGpad = NWG + 4;      // LDS row pitch (16B aligned, conflict-free)
  const int m0 = (wave % tilesM) * 16;
  const int ntl = (wave / tilesM) * 16;  // local n offset of this wave's tile
  const int n0 = blockIdx.x * NWG;       // workgroup base column

  // ---- stage X tile (Cin rows x NWG cols) into LDS ----
  const int quadsPerRow = NWG >> 2;
  const int totQuads = Cin * quadsPerRow;  // always a multiple of 256
#if USE_ASYNC_LDS
  const unsigned ldsbase = (unsigned)(unsigned long long)(void*)&Xs[0];
#endif
  for (int i = tid; i < totQuads; i += 256) {
    const int c = i / quadsPerRow;
    const int q = i - c * quadsPerRow;
    const int cs_ = c >> 6;
    const float* xp = cs_ == 0 ? X0 : cs_ == 1 ? X1 : cs_ == 2 ? X2 : X3;
    const float* ga =
        xp + (size_t)b * 64 * HWC + (size_t)(c & 63) * HWC + n0 + q * 4;
#if USE_ASYNC_LDS
    const unsigned lo = ldsbase + (unsigned)((c * NWGpad + q * 4) * 4);
    asm volatile("global_load_async_to_lds_b128 %0, %1, off" ::"v"(lo), "v"(ga)
                 : "memory");
#else
    *(float4*)(&Xs[c * NWGpad + q * 4]) = *(const float4*)ga;
#endif
  }
#if USE_ASYNC_LDS
  asm volatile("s_wait_asynccnt 0x0" ::: "memory");
#endif
  __syncthreads();

  v8f acc = {0.f, 0.f, 0.f, 0.f, 0.f, 0.f, 0.f, 0.f};
  for (int k0 = 0; k0 < Cin; k0 += 4) {
    const int kg = k0 + 2 * half;
    const v2f a = *(const v2f*)(We + (size_t)(m0 + ln) * Cin + kg);  // b64
    v2f bb;
    bb.x = Xs[kg * NWGpad + ntl + ln];
    bb.y = Xs[(kg + 1) * NWGpad + ntl + ln];
    acc = wmma_f32_k4(a, bb, acc);
  }
#pragma unroll
  for (int r = 0; r < 8; ++r) {
    const int m = m0 + r + 8 * half;
    const int n = n0 + ntl + ln;
    float y = acc[r] + (be ? be[m] : 0.f);
    if (s1) y = s1[m] * y + b1[m];
    if (relu1) y = fmaxf(y, 0.f);
    if (s2) {
      y = s2[m] * y + b2[m];
      y = fmaxf(y, 0.f);
    }
    const size_t oidx = (size_t)(b * M + m) * HWC + n;
    if (resadd) y = fmaxf(y + resadd[oidx], 0.f);
    out[oidx] = y;
  }
}

// ---------------------------------------------------------------------------
// Criss-cross attention, pass 1a: per (b,w) column, EH = Q^T K (diag masked),
// online per-row (max, sumexp) -> mH, lH.  qk: rows 0..7 = q, 8..15 = k.
// ---------------------------------------------------------------------------
__global__ __launch_bounds__(256) void dcam_stats_h(const float* __restrict__ qk,
                                                    float* __restrict__ mH,
                                                    float* __restrict__ lH) {
  const int w = blockIdx.x, b = blockIdx.y;
  __shared__ float Qs[1024], Ks[1024];
  const int tid = threadIdx.x;
  for (int i = tid; i < 1024; i += 256) {
    int c = i >> 7, hh = i & 127;
    Qs[i] = qk[(size_t)(b * 16 + c) * HWC + hh * WDIM + w];
    Ks[i] = qk[(size_t)(b * 16 + 8 + c) * HWC + hh * WDIM + w];
  }
  __syncthreads();
  const int wave = tid >> 5, lane = tid & 31, half = lane >> 4, ln = lane & 15;
  const int h0 = wave * 16;
  float mr[8], lr[8];
#pragma unroll
  for (int r = 0; r < 8; ++r) { mr[r] = -1e30f; lr[r] = 0.f; }
  for (int kt = 0; kt < 8; ++kt) {
    const int K0 = kt * 16;
    v8f acc = {0.f, 0.f, 0.f, 0.f, 0.f, 0.f, 0.f, 0.f};
#pragma unroll
    for (int cs = 0; cs < 8; cs += 4) {
      v2f a, bb;
      a.x = Qs[(cs + 2 * half) * 128 + h0 + ln];
      a.y = Qs[(cs + 2 * half + 1) * 128 + h0 + ln];
      bb.x = Ks[(cs + 2 * half) * 128 + K0 + ln];
      bb.y = Ks[(cs + 2 * half + 1) * 128 + K0 + ln];
      acc = wmma_f32_k4(a, bb, acc);
    }
#pragma unroll
    for (int r = 0; r < 8; ++r) {
      float e = acc[r];
      const int hh = h0 + r + 8 * half, KK = K0 + ln;
      if (hh == KK) e = -1e30f;  // diagonal mask (exp -> 0)
      float mo = mr[r], mn = fmaxf(mo, e);
      lr[r] = lr[r] * __expf(mo - mn) + __expf(e - mn);
      mr[r] = mn;
    }
  }
#pragma unroll
  for (int r = 0; r < 8; ++r) {
    float m = mr[r], l = lr[r];
    for (int off = 1; off < 16; off <<= 1) {
      float m2 = __shfl_xor(m, off, 32);
      float l2 = __shfl_xor(l, off, 32);
      float mn = fmaxf(m, m2);
      l = l * __expf(m - mn) + l2 * __expf(m2 - mn);
      m = mn;
    }
    if (ln == 0) {
      const int hh = h0 + r + 8 * half;
      mH[(size_t)b * HWC + hh * WDIM + w] = m;
      lH[(size_t)b * HWC + hh * WDIM + w] = l;
    }
  }
}

// ---------------------------------------------------------------------------
// Pass 1b: per (b,h) row, EW = Q^T K, stats; combine with (mH,lH) -> (mF,lF).
// ---------------------------------------------------------------------------
__global__ __launch_bounds__(256) void dcam_stats_w(
    const float* __restrict__ qk, const float* __restrict__ mH,
    const float* __restrict__ lH, float* __restrict__ mF,
    float* __restrict__ lF) {
  const int h = blockIdx.x, b = blockIdx.y;
  __shared__ float Qs[1024], Ks[1024];
  const int tid = threadIdx.x;
  for (int i = tid; i < 1024; i += 256) {
    int c = i >> 7, ww = i & 127;
    Qs[i] = qk[(size_t)(b * 16 + c) * HWC + h * WDIM + ww];
    Ks[i] = qk[(size_t)(b * 16 + 8 + c) * HWC + h * WDIM + ww];
  }
  __syncthreads();
  const int wave = tid >> 5, lane = tid & 31, half = lane >> 4, ln = lane & 15;
  const int w0 = wave * 16;
  float mr[8], lr[8];
#pragma unroll
  for (int r = 0; r < 8; ++r) { mr[r] = -1e30f; lr[r] = 0.f; }
  for (int kt = 0; kt < 8; ++kt) {
    const int K0 = kt * 16;
    v8f acc = {0.f, 0.f, 0.f, 0.f, 0.f, 0.f, 0.f, 0.f};
#pragma unroll
    for (int cs = 0; cs < 8; cs += 4) {
      v2f a, bb;
      a.x = Qs[(cs + 2 * half) * 128 + w0 + ln];
      a.y = Qs[(cs + 2 * half + 1) * 128 + w0 + ln];
      bb.x = Ks[(cs + 2 * half) * 128 + K0 + ln];
      bb.y = Ks[(cs + 2 * half + 1) * 128 + K0 + ln];
      acc = wmma_f32_k4(a, bb, acc);
    }
#pragma unroll
    for (int r = 0; r < 8; ++r) {
      float e = acc[r];
      float mo = mr[r], mn = fmaxf(mo, e);
      lr[r] = lr[r] * __expf(mo - mn) + __expf(e - mn);
      mr[r] = mn;
    }
  }
#pragma unroll
  for (int r = 0; r < 8; ++r) {
    float m = mr[r], l = lr[r];
    for (int off = 1; off < 16; off <<= 1) {
      float m2 = __shfl_xor(m, off, 32);
      float l2 = __shfl_xor(l, off, 32);
      float mn = fmaxf(m, m2);
      l = l * __expf(m - mn) + l2 * __expf(m2 - mn);
      m = mn;
    }
    if (ln == 0) {
      const size_t idx = (size_t)b * HWC + h * WDIM + (w0 + r + 8 * half);
      float mh = mH[idx], lh = lH[idx];
      float mn = fmaxf(m, mh);
      lF[idx] = lh * __expf(mh - mn) + l * __expf(m - mn);
      mF[idx] = mn;
    }
  }
}

// ---------------------------------------------------------------------------
// Pass 2a: per (b,w): recompute EH per 16-K strip, a=exp(e-m)/l (diag->0) in
// LDS, accumulate outH = V * a^T  (V = v[:, K, w]).  Writes O.
// ---------------------------------------------------------------------------
__global__ __launch_bounds__(256) void dcam_apply_h(
    const float* __restrict__ qk, const float* __restrict__ v,
    const float* __restrict__ mF, const float* __restrict__ lF,
    float* __restrict__ O) {
  const int w = blockIdx.x, b = blockIdx.y;
  __shared__ float Qs[1024], Ks[1024], Vs[8192], As[2048];
  const int tid = threadIdx.x;
  for (int i = tid; i < 1024; i += 256) {
    int c = i >> 7, hh = i & 127;
    Qs[i] = qk[(size_t)(b * 16 + c) * HWC + hh * WDIM + w];
    Ks[i] = qk[(size_t)(b * 16 + 8 + c) * HWC + hh * WDIM + w];
  }
  for (int i = tid; i < 8192; i += 256) {
    int c = i >> 7, K = i & 127;
    Vs[i] = v[(size_t)(b * 64 + c) * HWC + K * WDIM + w];
  }
  __syncthreads();
  const int wave = tid >> 5, lane = tid & 31, half = lane >> 4, ln = lane & 15;
  const int h0 = wave * 16;  // this wave's EH row strip
  float mrow[8], linv[8];
#pragma unroll
  for (int r = 0; r < 8; ++r) {
    const size_t idx = (size_t)b * HWC + (h0 + r + 8 * half) * WDIM + w;
    mrow[r] = mF[idx];
    linv[r] = 1.0f / lF[idx];
  }
  const int ct = (wave & 3) * 16;  // output c tile
  const int hg = (wave >> 2) * 4;  // output h tile group (4 tiles)
  v8f oacc[4];
#pragma unroll
  for (int j = 0; j < 4; ++j)
    oacc[j] = (v8f){0.f, 0.f, 0.f, 0.f, 0.f, 0.f, 0.f, 0.f};

  for (int kt = 0; kt < 8; ++kt) {
    const int K0 = kt * 16;
    v8f acc = {0.f, 0.f, 0.f, 0.f, 0.f, 0.f, 0.f, 0.f};
#pragma unroll
    for (int cs = 0; cs < 8; cs += 4) {
      v2f a, bb;
      a.x = Qs[(cs + 2 * half) * 128 + h0 + ln];
      a.y = Qs[(cs + 2 * half + 1) * 128 + h0 + ln];
      bb.x = Ks[(cs + 2 * half) * 128 + K0 + ln];
      bb.y = Ks[(cs + 2 * half + 1) * 128 + K0 + ln];
      acc = wmma_f32_k4(a, bb, acc);
    }
#pragma unroll
    for (int r = 0; r < 8; ++r) {
      const int hh = h0 + r + 8 * half, KK = K0 + ln;
      float aval = (hh == KK) ? 0.f : __expf(acc[r] - mrow[r]) * linv[r];
      As[ln * 128 + hh] = aval;  // As[Klocal][h]  (== a^T B-fragment source)
    }
    __syncthreads();
#pragma unroll
    for (int j = 0; j < 4; ++j) {
      const int n0 = (hg + j) * 16;
#pragma unroll
      for (int ks = 0; ks < 16; ks += 4) {
        v2f a, bb;
        a.x = Vs[(ct + ln) * 128 + K0 + ks + 2 * half];
        a.y = Vs[(ct + ln) * 128 + K0 + ks + 2 * half + 1];
        bb.x = As[(ks + 2 * half) * 128 + n0 + ln];
        bb.y = As[(ks + 2 * half + 1) * 128 + n0 + ln];
        oacc[j] = wmma_f32_k4(a, bb, oacc[j]);
      }
    }
    __syncthreads();
  }
#pragma unroll
  for (int j = 0; j < 4; ++j) {
    const int n0 = (hg + j) * 16;
#pragma unroll
    for (int r = 0; r < 8; ++r) {
      const int c = ct + r + 8 * half;
      const int hh = n0 + ln;
      O[(size_t)(b * 64 + c) * HWC + hh * WDIM + w] = oacc[j][r];
    }
  }
}

// ---------------------------------------------------------------------------
// Pass 2b: per (b,h): recompute EW, a=exp(e-m)/l, outW = V * a^T
// (V = v[:, h, K]).  Epilogue: t = gamma*(O + outW) + t   (in place).
// ---------------------------------------------------------------------------
__global__ __launch_bounds__(256) void dcam_apply_w(
    const float* __restrict__ qk, const float* __restrict__ v,
    const float* __restrict__ mF, const float* __restrict__ lF,
    const float* __restrict__ O, const float* __restrict__ gptr,
    float* __restrict__ t) {
  const int h = blockIdx.x, b = blockIdx.y;
  __shared__ float Qs[1024], Ks[1024], Vs[8192], As[2048];
  const int tid = threadIdx.x;
  for (int i = tid; i < 1024; i += 256) {
    int c = i >> 7, ww = i & 127;
    Qs[i] = qk[(size_t)(b * 16 + c) * HWC + h * WDIM + ww];
    Ks[i] = qk[(size_t)(b * 16 + 8 + c) * HWC + h * WDIM + ww];
  }
  for (int i = tid; i < 8192; i += 256) {
    int c = i >> 7, K = i & 127;
    Vs[i] = v[(size_t)(b * 64 + c) * HWC + h * WDIM + K];
  }
  __syncthreads();
  const int wave = tid >> 5, lane = tid & 31, half = lane >> 4, ln = lane & 15;
  const int w0 = wave * 16;
  float mrow[8], linv[8];
#pragma unroll
  for (int r = 0; r < 8; ++r) {
    const size_t idx = (size_t)b * HWC + h * WDIM + (w0 + r + 8 * half);
    mrow[r] = mF[idx];
    linv[r] = 1.0f / lF[idx];
  }
  const int ct = (wave & 3) * 16;
  const int hg = (wave >> 2) * 4;
  v8f oacc[4];
#pragma unroll
  for (int j = 0; j < 4; ++j)
    oacc[j] = (v8f){0.f, 0.f, 0.f, 0.f, 0.f, 0.f, 0.f, 0.f};

  for (int kt = 0; kt < 8; ++kt) {
    const int K0 = kt * 16;
    v8f acc = {0.f, 0.f, 0.f, 0.f, 0.f, 0.f, 0.f, 0.f};
#pragma unroll
    for (int cs = 0; cs < 8; cs += 4) {
      v2f a, bb;
      a.x = Qs[(cs + 2 * half) * 128 + w0 + ln];
      a.y = Qs[(cs + 2 * half + 1) * 128 + w0 + ln];
      bb.x = Ks[(cs + 2 * half) * 128 + K0 + ln];
      bb.y = Ks[(cs + 2 * half + 1) * 128 + K0 + ln];
      acc = wmma_f32_k4(a, bb, acc);
    }
#pragma unroll
    for (int r = 0; r < 8; ++r) {
      const int wwr = w0 + r + 8 * half;
      As[ln * 128 + wwr] = __expf(acc[r] - mrow[r]) * linv[r];
    }
    __syncthreads();
#pragma unroll
    for (int j = 0; j < 4; ++j) {
      const int n0 = (hg + j) * 16;
#pragma unroll
      for (int ks = 0; ks < 16; ks += 4) {
        v2f a, bb;
        a.x = Vs[(ct + ln) * 128 + K0 + ks + 2 * half];
        a.y = Vs[(ct + ln) * 128 + K0 + ks + 2 * half + 1];
        bb.x = As[(ks + 2 * half) * 128 + n0 + ln];
        bb.y = As[(ks + 2 * half + 1) * 128 + n0 + ln];
        oacc[j] = wmma_f32_k4(a, bb, oacc[j]);
      }
    }
    __syncthreads();
  }
  const float g = gptr[0];
#pragma unroll
  for (int j = 0; j < 4; ++j) {
    const int n0 = (hg + j) * 16;
#pragma unroll
    for (int r = 0; r < 8; ++r) {
      const int c = ct + r + 8 * half;
      const int ww = n0 + ln;
      const size_t idx = (size_t)(b * 64 + c) * HWC + h * WDIM + ww;
      t[idx] = g * (O[idx] + oacc[j][r]) + t[idx];
    }
  }
}

// ---------------------------------------------------------------------------
// Host orchestration
// ---------------------------------------------------------------------------
extern "C" void kernel_launch(void* const* d_in, const int* in_sizes, int n_in,
                              void* d_out, int out_size, void* d_ws,
                              size_t ws_size, hipStream_t stream) {
  auto F = [&](int i) { return (const float*)d_in[i]; };
  const float *x, *w_b0c0, *s_b0c0, *b_b0c0, *s_b0bn, *b_b0bn;
  const float *w_b1c0, *s_b1c0, *b_b1c0, *s_b1bn, *b_b1bn, *w_b1c1, *s_b1c1, *b_b1c1;
  const float *w_b2c0, *s_b2c0, *b_b2c0, *s_b2bn0, *b_b2bn0, *w_b2c1, *s_b2c1,
      *b_b2c1, *s_b2bn1, *b_b2bn1;
  const float *w_b3c0, *s_b3c0, *b_b3c0, *s_b3bn0, *b_b3bn0, *w_b3c1, *s_b3c1,
      *b_b3c1, *s_b3bn1, *b_b3bn1, *w_b3c2, *s_b3c2, *b_b3c2;
  const float *w_cat, *s_cat, *b_cat, *w_res, *s_res, *b_res;
  const float *wq, *bq, *wk, *bk, *wv, *bv, *gamma;

  if (in_sizes[0] == 8 * 64 * 128 * 128) {
    // Plain dict insertion-order flattening: x first, then params.
    x = F(0);
    w_b0c0 = F(1); s_b0c0 = F(2); b_b0c0 = F(3);
    s_b0bn = F(4); b_b0bn = F(5);
    w_b1c0 = F(6); s_b1c0 = F(7); b_b1c0 = F(8);
    s_b1bn = F(9); b_b1bn = F(10);
    w_b1c1 = F(11); s_b1c1 = F(12); b_b1c1 = F(13);
    w_b2c0 = F(14); s_b2c0 = F(15); b_b2c0 = F(16);
    s_b2bn0 = F(17); b_b2bn0 = F(18);
    w_b2c1 = F(19); s_b2c1 = F(20); b_b2c1 = F(21);
    s_b2bn1 = F(22); b_b2bn1 = F(23);
    w_b3c0 = F(24); s_b3c0 = F(25); b_b3c0 = F(26);
    s_b3bn0 = F(27); b_b3bn0 = F(28);
    w_b3c1 = F(29); s_b3c1 = F(30); b_b3c1 = F(31);
    s_b3bn1 = F(32); b_b3bn1 = F(33);
    w_b3c2 = F(34); s_b3c2 = F(35); b_b3c2 = F(36);
    w_cat = F(37); s_cat = F(38); b_cat = F(39);
    w_res = F(40); s_res = F(41); b_res = F(42);
    wq = F(43); bq = F(44); wk = F(45); bk = F(46); wv = F(47); bv = F(48);
    gamma = F(49);
  } else {
    // jax pytree flattening (sorted dict keys): params first, x last.
    b_b0bn = F(0); s_b0bn = F(1);
    b_b0c0 = F(2); s_b0c0 = F(3); w_b0c0 = F(4);
    b_b1bn = F(5); s_b1bn = F(6);
    b_b1c0 = F(7); s_b1c0 = F(8); w_b1c0 = F(9);
    b_b1c1 = F(10); s_b1c1 = F(11); w_b1c1 = F(12);
    b_b2bn0 = F(13); s_b2bn0 = F(14);
    b_b2bn1 = F(15); s_b2bn1 = F(16);
    b_b2c0 = F(17); s_b2c0 = F(18); w_b2c0 = F(19);
    b_b2c1 = F(20); s_b2c1 = F(21); w_b2c1 = F(22);
    b_b3bn0 = F(23); s_b3bn0 = F(24);
    b_b3bn1 = F(25); s_b3bn1 = F(26);
    b_b3c0 = F(27); s_b3c0 = F(28); w_b3c0 = F(29);
    b_b3c1 = F(30); s_b3c1 = F(31); w_b3c1 = F(32);
    b_b3c2 = F(33); s_b3c2 = F(34); w_b3c2 = F(35);
    b_cat = F(36); s_cat = F(37); w_cat = F(38);
    b_res = F(39); s_res = F(40); w_res = F(41);
    bk = F(42); bq = F(43); bv = F(44); gamma = F(45);
    wk = F(46); wq = F(47); wv = F(48);
    x = F(49);
  }

  // ---- workspace carve-up (floats). ~245 MB total. ----
  const size_t ACT = (size_t)8 * 64 * HWC;  // one (B,64,H,W) activation
  float* ws = (float*)d_ws;
  float* t0 = ws;
  float* t1 = t0 + ACT;
  float* t2 = t1 + ACT;
  float* t3 = t2 + ACT;
  float* ua = t3 + ACT;  // GEMM temp, aliases attention O buffer
  float* ub = ua + ACT;  // GEMM temp, aliases res-branch output
  float* vbuf = ub + ACT;
  float* Obuf = ua;      // alias: free whenever a dcam runs
  float* resout = ub;    // alias: ub free after branch-3 GEMMs
  float* qkbuf = vbuf + ACT;  // (B,16,H,W)
  float* mH = qkbuf + (size_t)8 * 16 * HWC;
  float* lH = mH + (size_t)8 * HWC;
  float* mF = lH + (size_t)8 * HWC;
  float* lF = mF + (size_t)8 * HWC;
  float* Wqk = lF + (size_t)8 * HWC;  // 16x64
  float* beqk = Wqk + 16 * 64;
  float* Wslot = beqk + 16;  // up to 64x256
  float* beslot = Wslot + 64 * 256;
  (void)ws_size; (void)n_in; (void)out_size;

  const dim3 gridM64(512, 8), gridM16(128, 8), gridPix(128, 8);

  auto gemm = [&](const float* X0, const float* X1, const float* X2,
                  const float* X3, int Cin, int M, const float* We,
                  const float* be, const float* s1v, const float* b1v,
                  int relu1, const float* s2v, const float* b2v,
                  const float* res, float* out) {
    conv1x1_gemm<<<(M == 64 ? gridM64 : gridM16), 256, 0, stream>>>(
        X0, X1, X2, X3, Cin, M, We, be, s1v, b1v, relu1, s2v, b2v, res, out);
  };
  auto prep = [&](const float* W, const float* bias, const float* si,
                  const float* bi, float* We, float* be, int M, int Cin) {
    prep_weight<<<M, 256, 0, stream>>>(W, bias, si, bi, We, be, Cin);
  };
  auto dcam = [&](float* t) {
    gemm(t, nullptr, nullptr, nullptr, 64, 16, Wqk, beqk, nullptr, nullptr, 0,
         nullptr, nullptr, nullptr, qkbuf);  // q|k
    gemm(t, nullptr, nullptr, nullptr, 64, 64, wv, bv, nullptr, nullptr, 0,
         nullptr, nullptr, nullptr, vbuf);   // v
    dcam_stats_h<<<gridPix, 256, 0, stream>>>(qkbuf, mH, lH);
    dcam_stats_w<<<gridPix, 256, 0, stream>>>(qkbuf, mH, lH, mF, lF);
    dcam_apply_h<<<gridPix, 256, 0, stream>>>(qkbuf, vbuf, mF, lF, Obuf);
    dcam_apply_w<<<gridPix, 256, 0, stream>>>(qkbuf, vbuf, mF, lF, Obuf, gamma,
                                              t);  // t in place
  };

  // shared attention qk weight pack (q -> rows 0..7, k -> rows 8..15)
  prep(wq, bq, nullptr, nullptr, Wqk, beqk, 8, 64);
  prep(wk, bk, nullptr, nullptr, Wqk + 8 * 64, beqk + 8, 8, 64);

  // branch 0: relu(bn(relu(s*W0x+b)))  -> dcam
  gemm(x, nullptr, nullptr, nullptr, 64, 64, w_b0c0, nullptr, s_b0c0, b_b0c0, 1,
       s_b0bn, b_b0bn, nullptr, t0);
  dcam(t0);

  // branch 1
  gemm(x, nullptr, nullptr, nullptr, 64, 64, w_b1c0, nullptr, s_b1c0, b_b1c0, 1,
       nullptr, nullptr, nullptr, ua);
  prep(w_b1c1, nullptr, s_b1bn, b_b1bn, Wslot, beslot, 64, 64);
  gemm(ua, nullptr, nullptr, nullptr, 64, 64, Wslot, beslot, s_b1c1, b_b1c1, 1,
       nullptr, nullptr, nullptr, t1);
  dcam(t1);

  // branch 2
  gemm(x, nullptr, nullptr, nullptr, 64, 64, w_b2c0, nullptr, s_b2c0, b_b2c0, 1,
       nullptr, nullptr, nullptr, ua);
  prep(w_b2c1, nullptr, s_b2bn0, b_b2bn0, Wslot, beslot, 64, 64);
  gemm(ua, nullptr, nullptr, nullptr, 64, 64, Wslot, beslot, s_b2c1, b_b2c1, 1,
       s_b2bn1, b_b2bn1, nullptr, t2);
  dcam(t2);

  // branch 3
  gemm(x, nullptr, nullptr, nullptr, 64, 64, w_b3c0, nullptr, s_b3c0, b_b3c0, 1,
       nullptr, nullptr, nullptr, ua);
  prep(w_b3c1, nullptr, s_b3bn0, b_b3bn0, Wslot, beslot, 64, 64);
  gemm(ua, nullptr, nullptr, nullptr, 64, 64, Wslot, beslot, s_b3c1, b_b3c1, 1,
       nullptr, nullptr, nullptr, ub);
  prep(w_b3c2, nullptr, s_b3bn1, b_b3bn1, Wslot, beslot, 64, 64);
  gemm(ub, nullptr, nullptr, nullptr, 64, 64, Wslot, beslot, s_b3c2, b_b3c2, 1,
       nullptr, nullptr, nullptr, t3);
  dcam(t3);

  // residual branch and final concat GEMM: out = relu(basic(cat) + basic(x))
  gemm(x, nullptr, nullptr, nullptr, 64, 64, w_res, nullptr, s_res, b_res, 1,
       nullptr, nullptr, nullptr, resout);
  gemm(t0, t1, t2, t3, 256, 64, w_cat, nullptr, s_cat, b_cat, 1, nullptr,
       nullptr, resout, (float*)d_out);
}